// GraphODEFunc_781684048056
// MI455X (gfx1250) — compile-verified
//
#include <hip/hip_runtime.h>
#include <hip/hip_bf16.h>

// Problem constants (match reference)
#define NN 12000   // num nodes
#define HH 32      // hidden dim
#define TT 50      // time steps

typedef __attribute__((ext_vector_type(2))) float v2f;
typedef __attribute__((ext_vector_type(8))) float v8f;

// -----------------------------------------------------------------------------
// Kernel 1: support[n,h] = ([z | a_t] @ W)[n,h]
//   a_idx = clip(trunc(t[0]*(T-1)), 0, T-1); a_t[n] = treatments[n, a_idx, 0]
// Tiny (25 MFLOP); plain fp32 VALU is fine.
// -----------------------------------------------------------------------------
__global__ __launch_bounds__(256) void support_kernel(
    const float* __restrict__ t,
    const float* __restrict__ z,          // [N, H]
    const float* __restrict__ treatments, // [N, T]
    const float* __restrict__ W,          // [H+1, H]
    float* __restrict__ support)          // [N, H]  (workspace)
{
    int idx = blockIdx.x * blockDim.x + threadIdx.x;
    if (idx >= NN * HH) return;
    int n = idx >> 5;       // row
    int h = idx & (HH - 1); // col

    int a_idx = (int)(t[0] * (float)(TT - 1)); // trunc toward zero, t in [0,1)
    a_idx = min(max(a_idx, 0), TT - 1);
    float a_n = treatments[n * TT + a_idx];

    float acc = a_n * W[HH * HH + h];   // last row of W (H+1 rows)
    const float* zrow = z + (size_t)n * HH;
#pragma unroll
    for (int j = 0; j < HH; ++j)
        acc = fmaf(zrow[j], W[j * HH + h], acc);
    support[(size_t)n * HH + h] = acc;
}

// -----------------------------------------------------------------------------
// Kernel 2: out = relu(adj @ support + b) via V_WMMA_F32_16X16X4_F32
// One wave32 per 16-row tile of the output (750 tiles ~= one wave per SIMD).
//
// Fragment layouts (CDNA5 ISA 7.12.2, 32-bit, wave32):
//   A 16x4 (MxK):  lane L, VGPR g -> A[L%16][2*(L/16) + g]
//   B 4x16 (KxN):  lane L, VGPR g -> B[2*(L/16) + g][L%16]
//   C/D 16x16:     lane L, VGPR r -> D[r + 8*(L/16)][L%16]
//
// adj (576 MB) is streamed once: non-temporal loads keep the 1.5 MB `support`
// table hot in the 192 MB L2. global_prefetch_b8 (no LOADcnt cost) runs
// ~16 KB/row ahead on adj (>1000 cycles of lead vs HBM latency) and 2 chunks
// ahead on support (one 128B line per lane = one support row).
// -----------------------------------------------------------------------------
__global__ __launch_bounds__(32) void aggregate_kernel(
    const float* __restrict__ adj,     // [N, N]
    const float* __restrict__ support, // [N, H]
    const float* __restrict__ bias,    // [H]
    float* __restrict__ out)           // [N, H]
{
    const int lane  = threadIdx.x;        // 0..31, full wave active (EXEC all 1s)
    const int row0  = blockIdx.x * 16;    // output row tile base
    const int lrow  = lane & 15;          // A row / B col / D col within tile
    const int khalf = (lane >> 4) << 1;   // 0 or 2 : K sub-offset for this lane half

    v8f acc0 = {}; // output cols  0..15
    v8f acc1 = {}; // output cols 16..31

    // Per-lane base of this lane's adj row; row byte stride 48000 (8B aligned),
    // khalf even -> v2f (b64) loads are 8B aligned.
    const float* arow = adj + (size_t)(row0 + lrow) * NN;

    for (int k = 0; k < NN; k += 32) {
        // Deep prefetch of this lane's adj row: 4096 floats (16 KB, ~128 chunk
        // iterations) ahead -- speculative, dropped silently past the end.
        __builtin_prefetch(arow + k + 4096, 0, 1);
        // Prefetch support rows k+64..k+95 (2 chunks ahead): one row = one
        // 128B line per lane, 32 contiguous lines.
        __builtin_prefetch(support + (size_t)(k + 64 + lane) * HH, 0, 1);
#pragma unroll
        for (int kk = 0; kk < 32; kk += 4) {
            const int kb = k + kk + khalf;           // this lane's K pair base
            // A fragment: adj[row0+lrow][kb], adj[row0+lrow][kb+1]  (streamed, NT)
            v2f a = __builtin_nontemporal_load((const v2f*)(arow + kb));
            // B fragments from support (L2-resident)
            const float* s0 = support + (size_t)kb * HH;        // row kb
            const float* s1 = s0 + HH;                          // row kb+1
            v2f b0, b1;
            b0.x = s0[lrow];        b0.y = s1[lrow];            // cols 0..15
            b1.x = s0[lrow + 16];   b1.y = s1[lrow + 16];       // cols 16..31
            // D = A*B + C   (8 args: neg_a, A, neg_b, B, c_mod, C, reuse_a, reuse_b)
            acc0 = __builtin_amdgcn_wmma_f32_16x16x4_f32(
                false, a, false, b0, (short)0, acc0, false, false);
            acc1 = __builtin_amdgcn_wmma_f32_16x16x4_f32(
                false, a, false, b1, (short)0, acc1, false, false);
        }
    }

    // Epilogue: bias + relu + store per C/D layout.
    const float bc0 = bias[lrow];
    const float bc1 = bias[lrow + 16];
    const int rbase = row0 + ((lane >> 4) << 3);   // +8 rows for upper lane half
#pragma unroll
    for (int r = 0; r < 8; ++r) {
        float v0 = fmaxf(acc0[r] + bc0, 0.0f);
        float v1 = fmaxf(acc1[r] + bc1, 0.0f);
        float* orow = out + (size_t)(rbase + r) * HH;
        orow[lrow]      = v0;
        orow[lrow + 16] = v1;
    }
}

// -----------------------------------------------------------------------------
// Launch. Inputs (setup_inputs order): t[1], z[N*H], adj[N*N],
// treatments[N*T], W[(H+1)*H], b[H]. Output: [N*H] fp32.
// -----------------------------------------------------------------------------
extern "C" void kernel_launch(void* const* d_in, const int* in_sizes, int n_in,
                              void* d_out, int out_size, void* d_ws, size_t ws_size,
                              hipStream_t stream) {
    const float* t          = (const float*)d_in[0];
    const float* z          = (const float*)d_in[1];
    const float* adj        = (const float*)d_in[2];
    const float* treatments = (const float*)d_in[3];
    const float* W          = (const float*)d_in[4];
    const float* b          = (const float*)d_in[5];
    float* out     = (float*)d_out;
    float* support = (float*)d_ws;   // N*H*4 = 1.5 MB scratch

    (void)in_sizes; (void)n_in; (void)out_size; (void)ws_size;

    // 1) support = [z | a_t] @ W
    {
        int total = NN * HH;
        int threads = 256;
        int blocks = (total + threads - 1) / threads;
        support_kernel<<<blocks, threads, 0, stream>>>(t, z, treatments, W, support);
    }
    // 2) out = relu(adj @ support + b), one wave per 16-row tile
    {
        aggregate_kernel<<<NN / 16, 32, 0, stream>>>(adj, support, b, out);
    }
}